// SE3Denoiser_48301202211282
// MI455X (gfx1250) — compile-verified
//
#include <hip/hip_runtime.h>
#include <hip/hip_bf16.h>
#include <math.h>

typedef float v2f  __attribute__((ext_vector_type(2)));
typedef float v8f  __attribute__((ext_vector_type(8)));
typedef unsigned int u32x4 __attribute__((ext_vector_type(4)));
typedef int   i32x4 __attribute__((ext_vector_type(4)));
typedef int   i32x8 __attribute__((ext_vector_type(8)));

#define BATCH   8
#define NN      1024
#define KNN_K   20
#define SAMP_K  40
#define K_TOT   60
#define REMN    (NN - KNN_K)      // 1004
#define INF_VAL 100000000.0f

// Flat output offsets (elements), outputs concatenated in reference return order.
#define O_SRC 0ull                 // edge_sources (8, 61440)
#define O_SNK 491520ull            // edge_sinks   (8, 61440)
#define O_REL 983040ull            // rel_pos      (8,1024,60,3)
#define O_NF0 2457600ull           // node_feat_0  (8192,1,1)
#define O_NF1 2465792ull           // node_feat_1  (8192,3,3)
#define O_NAN 2539520ull           // is_nan_node  (8,1024)

// ---------------------------------------------------------------------------
// Kernel 0: per-node prep. Zero NaN nodes' coords, emit CA, |CA|^2, nan flag,
// and the cheap per-node outputs (node_feat_0/1, is_nan).
// ---------------------------------------------------------------------------
__global__ __launch_bounds__(256) void prep_kernel(
    const float* __restrict__ coords,   // (B,N,4,3)
    const float* __restrict__ ts,       // (B,1)
    float* __restrict__ ca,             // (B,N,3)
    float* __restrict__ sq,             // (B,N)
    int*   __restrict__ nanflag,        // (B,N)
    float* __restrict__ out)
{
    int bn = blockIdx.x * blockDim.x + threadIdx.x;
    if (bn >= BATCH * NN) return;
    int b = bn >> 10;

    const float* c0 = coords + (size_t)bn * 12;
    float c[12];
    bool anynan = false;
#pragma unroll
    for (int q = 0; q < 12; ++q) {
        c[q] = c0[q];
        anynan = anynan || (c[q] != c[q]);   // NaN check
    }
    if (anynan) {
#pragma unroll
        for (int q = 0; q < 12; ++q) c[q] = 0.0f;
    }
    float cax = c[3], cay = c[4], caz = c[5];
    ca[bn * 3 + 0] = cax;
    ca[bn * 3 + 1] = cay;
    ca[bn * 3 + 2] = caz;
    sq[bn] = cax * cax + cay * cay + caz * caz;
    nanflag[bn] = anynan ? 1 : 0;

    out[O_NF0 + bn] = ts[b];
    float* nf = out + O_NF1 + (size_t)bn * 9;
    nf[0] = c[0]  - cax; nf[1] = c[1]  - cay; nf[2] = c[2]  - caz;  // Na - CA
    nf[3] = c[6]  - cax; nf[4] = c[7]  - cay; nf[5] = c[8]  - caz;  // C  - CA
    nf[6] = c[9]  - cax; nf[7] = c[10] - cay; nf[8] = c[11] - caz;  // O  - CA
    out[O_NAN + bn] = anynan ? 1.0f : 0.0f;
}

// ---------------------------------------------------------------------------
// Kernel 1: pairwise SQUARED distance matrix via V_WMMA_F32_16X16X4_F32.
// A row m = (x0,x1,x2,1), B col n = (-2y0,-2y1,-2y2,|y|^2):
//   (A*B)[m][n] = |y|^2 - 2 x.y ; add |x|^2 -> d^2  (sqrt elided: order-equiv).
// One wave owns a 16-row strip and 8 column tiles; A loaded once, 8 WMMAs.
// A layout (16x4 f32): lanes 0-15: v0=K0, v1=K1 ; lanes 16-31: v0=K2, v1=K3.
// D layout: VGPR r -> M = r + (lane>=16 ? 8:0), N = lane&15.
// ---------------------------------------------------------------------------
__global__ __launch_bounds__(256) void dist_wmma_kernel(
    const float* __restrict__ ca,       // (B,N,3) zeroed CA
    const float* __restrict__ sq,       // (B,N)
    const int*   __restrict__ nanflag,  // (B,N)
    float* __restrict__ dist)           // (B,N,N) squared distances
{
    const int lane = threadIdx.x & 31;
    const int wave = threadIdx.x >> 5;
    const int wid  = blockIdx.x * 8 + wave;   // 0 .. 4095  (B * 64 ti * 8 jg)
    const int b  = wid >> 9;
    const int ti = (wid >> 3) & 63;
    const int jg = wid & 7;
    const int i0 = ti << 4;

    const float* cab = ca + (size_t)b * NN * 3;
    const float* sqb = sq + (size_t)b * NN;
    const int*   nb  = nanflag + (size_t)b * NN;

    // --- A operand: fixed for all 8 column tiles ---
    const int mrow = i0 + (lane & 15);
    v2f a;
    if (lane < 16) { a.x = cab[mrow * 3 + 0]; a.y = cab[mrow * 3 + 1]; }
    else           { a.x = cab[mrow * 3 + 2]; a.y = 1.0f; }

    // Row-side |x|^2 and nan flags for this lane's 8 output rows.
    const int mbase = i0 + ((lane >> 4) << 3);
    float srow[8];
    int   nrow[8];
#pragma unroll
    for (int r = 0; r < 8; ++r) {
        srow[r] = sqb[mbase + r];
        nrow[r] = nb[mbase + r];
    }

    float* dbase = dist + (size_t)b * NN * NN;

#pragma unroll
    for (int jj = 0; jj < 8; ++jj) {
        const int j0   = ((jg << 3) + jj) << 4;
        const int ncol = j0 + (lane & 15);

        v2f bm;
        if (lane < 16) { bm.x = -2.0f * cab[ncol * 3 + 0]; bm.y = -2.0f * cab[ncol * 3 + 1]; }
        else           { bm.x = -2.0f * cab[ncol * 3 + 2]; bm.y = sqb[ncol]; }

        v8f c = {};
        c = __builtin_amdgcn_wmma_f32_16x16x4_f32(
            false, a, false, bm, (short)0, c, false, false);

        const int col     = j0 + (lane & 15);
        const int nan_col = nb[col];
#pragma unroll
        for (int r = 0; r < 8; ++r) {
            float d2 = fmaxf(c[r] + srow[r], 0.0f);
            if (nrow[r] && nan_col) d2 = INF_VAL;
            dbase[(size_t)(mbase + r) * NN + col] = d2;
        }
    }
}

// ---------------------------------------------------------------------------
// Bitonic sort of 1024 (key,val) pairs in LDS, 256 threads.
// ---------------------------------------------------------------------------
__device__ __forceinline__ void bitonic_sort_1024(float* key, int* val, bool asc)
{
    for (int k = 2; k <= NN; k <<= 1) {
        for (int j = k >> 1; j > 0; j >>= 1) {
            __syncthreads();
#pragma unroll
            for (int s = 0; s < NN / 256; ++s) {
                int t = threadIdx.x + s * 256;
                int p = t ^ j;
                if (p > t) {
                    bool dirUp = (((t & k) == 0) == asc);
                    float ka = key[t], kb = key[p];
                    if ((ka > kb) == dirUp) {
                        key[t] = kb; key[p] = ka;
                        int va = val[t]; val[t] = val[p]; val[p] = va;
                    }
                }
            }
        }
    }
    __syncthreads();
}

// ---------------------------------------------------------------------------
// Kernel 2: per-row sort + Gumbel top-k + edge emission. One block per row.
// The 4KB distance row is pulled into LDS by the Tensor Data Mover (TDM).
// ---------------------------------------------------------------------------
__global__ __launch_bounds__(256) void sort_topk_kernel(
    const float* __restrict__ dist,     // (B,N,N) squared distances
    const float* __restrict__ noise,    // (B,N,N-20)
    const float* __restrict__ ca,       // (B,N,3)
    float* __restrict__ out)
{
    __shared__ float key[NN];
    __shared__ int   val[NN];
    __shared__ int   knn_e[KNN_K];

    const int row = blockIdx.x;          // b*N + i
    const int b = row >> 10;
    const int i = row & (NN - 1);

    const float* drow = dist + (size_t)row * NN;
    const float* nrow = noise + (size_t)row * REMN;

    // --- TDM: async DMA of the 1024-float row into LDS key[] ---
    if ((threadIdx.x >> 5) == 0) {
        unsigned long long gaddr = (unsigned long long)(uintptr_t)drow;
        unsigned int lds_off = (unsigned int)(uintptr_t)&key[0];

        u32x4 g0;
        g0.x = 1u;                                         // count=1 (valid D#)
        g0.y = lds_off;                                    // lds_addr (bytes)
        g0.z = (unsigned int)(gaddr & 0xFFFFFFFFu);        // global_addr[31:0]
        g0.w = (unsigned int)((gaddr >> 32) & 0x1FFFFFFu)  // global_addr[56:32]
             | 0x80000000u;                                // type=2 ("image")

        i32x8 g1;
        g1[0] = 0x00020000;            // data_size=2 (4 bytes), no flags
        g1[1] = (int)(1024u << 16);    // tensor_dim0[15:0]=1024 in bits[63:48]
        g1[2] = (int)(1u << 16);       // tensor_dim0 hi=0 ; tensor_dim1[15:0]=1
        g1[3] = (int)(1024u << 16);    // tensor_dim1 hi=0 ; tile_dim0=1024
        g1[4] = 1;                     // tile_dim1=1 ; tile_dim2=0
        g1[5] = 1024;                  // tensor_dim0_stride[31:0]=1024
        g1[6] = 0;                     // stride hi / tensor_dim1_stride lo
        g1[7] = 0;

        i32x4 g2 = {0, 0, 0, 0};
        i32x4 g3 = {0, 0, 0, 0};
        i32x8 g4 = {0, 0, 0, 0, 0, 0, 0, 0};

        __builtin_amdgcn_tensor_load_to_lds(g0, g1, g2, g3, g4, 0);
        __builtin_amdgcn_s_wait_tensorcnt(0);
    }

    // Warm the noise row into cache while we sort.
    __builtin_prefetch(nrow + threadIdx.x * 4, 0, 0);

    for (int t = threadIdx.x; t < NN; t += 256) val[t] = t;
    __syncthreads();

    // Phase 1: ascending sort of squared distances -> full argsort order.
    bitonic_sort_1024(key, val, true);

    if (threadIdx.x < KNN_K) knn_e[threadIdx.x] = val[threadIdx.x];

    // Phase 2: perturbed log-probs for ranks [20,1024); pad to 1024 with -INF.
    // -3*log(d) == -1.5*log(d^2)
    float nk[NN / 256];
    int   nv[NN / 256];
#pragma unroll
    for (int s = 0; s < NN / 256; ++s) {
        int slot = threadIdx.x + s * 256;
        if (slot < REMN) {
            int r = slot + KNN_K;
            float lp = -1.5f * logf(key[r]);
            float g  = -logf(-logf(nrow[slot]));
            nk[s] = lp + g;
            nv[s] = val[r];
        } else {
            nk[s] = -INF_VAL;
            nv[s] = 0;
        }
    }
    __syncthreads();
#pragma unroll
    for (int s = 0; s < NN / 256; ++s) {
        int slot = threadIdx.x + s * 256;
        key[slot] = nk[s];
        val[slot] = nv[s];
    }

    // Descending sort on perturbed value -> first 40 = sampled edges.
    bitonic_sort_1024(key, val, false);

    if (threadIdx.x < K_TOT) {
        int k    = threadIdx.x;
        int sink = (k < KNN_K) ? knn_e[k] : val[k - KNN_K];
        size_t eidx = (size_t)row * K_TOT + k;
        out[O_SRC + eidx] = (float)i;
        out[O_SNK + eidx] = (float)sink;
        const float* cb = ca + (size_t)b * NN * 3;
        float rx = cb[i * 3 + 0] - cb[sink * 3 + 0];
        float ry = cb[i * 3 + 1] - cb[sink * 3 + 1];
        float rz = cb[i * 3 + 2] - cb[sink * 3 + 2];
        size_t rp = O_REL + eidx * 3;
        out[rp + 0] = rx;
        out[rp + 1] = ry;
        out[rp + 2] = rz;
    }
}

// ---------------------------------------------------------------------------
// Host launcher
// ---------------------------------------------------------------------------
extern "C" void kernel_launch(void* const* d_in, const int* in_sizes, int n_in,
                              void* d_out, int out_size, void* d_ws, size_t ws_size,
                              hipStream_t stream) {
    (void)in_sizes; (void)n_in; (void)out_size; (void)ws_size;

    const float* coords = (const float*)d_in[0];   // (8,1024,4,3)
    const float* ts     = (const float*)d_in[1];   // (8,1)
    const float* noise  = (const float*)d_in[2];   // (8,1024,1004)
    float* out = (float*)d_out;

    // Workspace layout
    char* ws = (char*)d_ws;
    float* dist = (float*)(ws);                                   // 8*1024*1024*4 = 33554432 B
    float* ca   = (float*)(ws + 33554432);                        // 8*1024*3*4    =    98304 B
    float* sq   = (float*)(ws + 33554432 + 98304);                // 8*1024*4      =    32768 B
    int*   nanf = (int*)  (ws + 33554432 + 98304 + 32768);        // 8*1024*4      =    32768 B

    // 0) per-node prep + cheap outputs
    prep_kernel<<<(BATCH * NN + 255) / 256, 256, 0, stream>>>(coords, ts, ca, sq, nanf, out);

    // 1) WMMA squared-distance matrix: 4096 waves, 8 column tiles each
    dist_wmma_kernel<<<512, 256, 0, stream>>>(ca, sq, nanf, dist);

    // 2) per-row sort + Gumbel top-k + edge outputs (TDM row load)
    sort_topk_kernel<<<BATCH * NN, 256, 0, stream>>>(dist, noise, ca, out);
}